// MMD_loss_2757369004769
// MI455X (gfx1250) — compile-verified
//
#include <hip/hip_runtime.h>
#include <hip/hip_bf16.h>
#include <math.h>

// MMD loss, MI455X (gfx1250), wave32.
//   x1, x2 : [4096, 256] fp32
// out = sqrt(m11 - 2*m12 + m22); m_ab = mean over 5 gammas of exp(-16*g*D).
// Fused fp32-WMMA GEMM + epilogue; Gram/distance matrices never hit memory.
// LDS-staged tiles (padded stride -> bank-conflict-free fragment loads),
// G11/G22 on lower-triangle blocks only (weight 2 off-diagonal).

typedef __attribute__((ext_vector_type(2))) float v2f;
typedef __attribute__((ext_vector_type(4))) float v4f;
typedef __attribute__((ext_vector_type(8))) float v8f;

#define N_ROWS 4096
#define KDIM   256
#define KC     32                   // K chunk staged in LDS
#define PS     36                   // padded LDS row stride (floats), 36%64 trick
#define TILES  (N_ROWS / 128)       // 32 block tiles per dimension
#define NPART  (TILES * TILES)      // 1024 partials per matrix

// ---------------------------------------------------------------- row norms
__global__ __launch_bounds__(256) void row_norms_kernel(
    const float* __restrict__ x, float* __restrict__ out)
{
    int r = blockIdx.x * blockDim.x + threadIdx.x;   // one row per thread
    if (r < N_ROWS) {
        const float* p = x + (size_t)r * KDIM;
        float s = 0.0f;
        #pragma unroll 4
        for (int k = 0; k < KDIM; k += 4) {
            v4f v = *(const v4f*)(p + k);
            s += v.x * v.x + v.y * v.y + v.z * v.z + v.w * v.w;
        }
        out[r] = s;
    }
}

// ------------------------------------------------- fused WMMA tile + epilog
// Block = 256 threads = 8 waves (wave32), waves in 2(M) x 4(N) grid.
// Wave tile: 64(M) x 32(N) = 4x2 tiles of 16x16 via V_WMMA_F32_16X16X4_F32.
// Block macro-tile: 128 x 128, K staged through LDS in chunks of 32.
__global__ __launch_bounds__(256) void mmd_tile_kernel(
    const float* __restrict__ A,   // [4096, 256] rows of left operand
    const float* __restrict__ B,   // [4096, 256] rows of right operand
    const float* __restrict__ an,  // row norms of A
    const float* __restrict__ bn,  // row norms of B
    float* __restrict__ partials,  // one partial sum per block
    int symmetric)
{
    __shared__ float sA[128 * PS];         // 18432 B
    __shared__ float sB[128 * PS];         // 18432 B

    const int tid  = threadIdx.x;
    const int bi   = blockIdx.y;   // M block tile
    const int bj   = blockIdx.x;   // N block tile
    const int slot = bi * gridDim.x + bj;

    if (symmetric && bj > bi) {    // upper triangle: contribute 0, skip work
        if (tid == 0) partials[slot] = 0.0f;
        return;
    }
    const float weight = (symmetric && bj < bi) ? 2.0f : 1.0f;

    const int wave = tid >> 5;
    const int lane = tid & 31;
    const int l15  = lane & 15;
    const int hi   = lane >> 4;   // 0: lanes 0-15, 1: lanes 16-31
    const int kh   = hi << 1;     // K sub-offset {0,2} per ISA 16x4 layout

    const int waveM  = wave & 1;             // 2 waves along M
    const int waveN  = wave >> 1;            // 4 waves along N
    const int m_base = bi * 128 + waveM * 64;
    const int n_base = bj * 128 + waveN * 32;

    // Staging map: chunk row = 32 floats = 8 float4; 8 threads/row, 32 rows
    // per pass, 4 passes for 128 rows. Consecutive lanes -> consecutive
    // float4s: fully coalesced global_load_b128.
    const int c4 = (tid & 7) * 4;            // float column within chunk
    const int r0 = tid >> 3;                 // 0..31
    const float* gA = A + (size_t)(bi * 128 + r0) * KDIM + c4;
    const float* gB = B + (size_t)(bj * 128 + r0) * KDIM + c4;
    float* stA = sA + r0 * PS + c4;
    float* stB = sB + r0 * PS + c4;

    // Fragment bases in LDS (A 16x4 / B 4x16 per ISA layout: lane -> row
    // l15 (+tile offset), cols k+kh..k+kh+1 -> contiguous b64, banks
    // conflict-free thanks to PS=36).
    const float* ldA = sA + (waveM * 64 + l15) * PS + kh;
    const float* ldB = sB + (waveN * 32 + l15) * PS + kh;

    v8f cc[4][2];
    #pragma unroll
    for (int tm = 0; tm < 4; ++tm)
        #pragma unroll
        for (int tn = 0; tn < 2; ++tn)
            cc[tm][tn] = (v8f){};

    for (int kc = 0; kc < KDIM; kc += KC) {
        __syncthreads();                     // previous chunk fully consumed
        #pragma unroll
        for (int p = 0; p < 4; ++p) {
            *(v4f*)(stA + p * 32 * PS) = *(const v4f*)(gA + (size_t)p * 32 * KDIM + kc);
            *(v4f*)(stB + p * 32 * PS) = *(const v4f*)(gB + (size_t)p * 32 * KDIM + kc);
        }
        __syncthreads();                     // chunk visible to all waves

        #pragma unroll
        for (int k = 0; k < KC; k += 4) {
            v2f a[4], b[2];
            #pragma unroll
            for (int tm = 0; tm < 4; ++tm)
                a[tm] = *(const v2f*)(ldA + tm * 16 * PS + k);
            #pragma unroll
            for (int tn = 0; tn < 2; ++tn)
                b[tn] = *(const v2f*)(ldB + tn * 16 * PS + k);
            // (neg_a, A, neg_b, B, c_mod, C, reuse_a, reuse_b)
            #pragma unroll
            for (int tm = 0; tm < 4; ++tm)
                #pragma unroll
                for (int tn = 0; tn < 2; ++tn)
                    cc[tm][tn] = __builtin_amdgcn_wmma_f32_16x16x4_f32(
                        false, a[tm], false, b[tn], (short)0, cc[tm][tn],
                        false, false);
        }
    }

    // coefs = sqrt(256) * 1/(2*b^2), b in {1,2,5,10,20}
    const float coef0 = 8.0f, coef1 = 2.0f, coef2 = 0.32f,
                coef3 = 0.08f, coef4 = 0.02f;

    // C/D layout: element r of v8f, lane -> M = r + 8*hi, N = l15
    float acc = 0.0f;
    #pragma unroll
    for (int tm = 0; tm < 4; ++tm) {
        const int mrow0 = m_base + tm * 16 + hi * 8;
        #pragma unroll
        for (int tn = 0; tn < 2; ++tn) {
            const float bno = bn[n_base + tn * 16 + l15];
            v8f c = cc[tm][tn];
            #pragma unroll
            for (int r = 0; r < 8; ++r) {
                float d = an[mrow0 + r] + bno - 2.0f * c[r];
                d = fmaxf(d, 0.0f);
                acc += __expf(-coef0 * d);
                acc += __expf(-coef1 * d);
                acc += __expf(-coef2 * d);
                acc += __expf(-coef3 * d);
                acc += __expf(-coef4 * d);
            }
        }
    }

    // Deterministic block reduction (reuse sA; no float atomics).
    __syncthreads();
    sA[tid] = acc;
    __syncthreads();
    #pragma unroll
    for (int off = 128; off > 0; off >>= 1) {
        if (tid < off) sA[tid] += sA[tid + off];
        __syncthreads();
    }
    if (tid == 0)
        partials[slot] = weight * sA[0];
}

// ------------------------------------------------------------- final reduce
__global__ __launch_bounds__(256) void mmd_finalize_kernel(
    const float* __restrict__ partials, float* __restrict__ out)
{
    __shared__ float red[256];
    const int t = threadIdx.x;
    float sums[3];
    for (int m = 0; m < 3; ++m) {
        float s = 0.0f;
        for (int i = t; i < NPART; i += 256) s += partials[m * NPART + i];
        red[t] = s;
        __syncthreads();
        #pragma unroll
        for (int off = 128; off > 0; off >>= 1) {
            if (t < off) red[t] += red[t + off];
            __syncthreads();
        }
        sums[m] = red[0];
        __syncthreads();
    }
    if (t == 0) {
        const float inv = 1.0f / (4096.0f * 4096.0f * 5.0f);
        float m11 = sums[0] * inv;
        float m22 = sums[1] * inv;
        float m12 = sums[2] * inv;
        out[0] = sqrtf(fmaxf(m11 - 2.0f * m12 + m22, 0.0f));
    }
}

// ---------------------------------------------------------------- launcher
extern "C" void kernel_launch(void* const* d_in, const int* in_sizes, int n_in,
                              void* d_out, int out_size, void* d_ws, size_t ws_size,
                              hipStream_t stream) {
    const float* x1 = (const float*)d_in[0];
    const float* x2 = (const float*)d_in[1];
    float* out = (float*)d_out;
    float* ws  = (float*)d_ws;

    // ws layout (floats): [0 .. 3*NPART) block partials (G11, G22, G12),
    //                     [3*NPART .. +4096) norms of x1,
    //                     [3*NPART+4096 .. +4096) norms of x2.
    float* partials = ws;
    float* n1 = ws + 3 * NPART;
    float* n2 = ws + 3 * NPART + N_ROWS;

    row_norms_kernel<<<dim3(N_ROWS / 256), dim3(256), 0, stream>>>(x1, n1);
    row_norms_kernel<<<dim3(N_ROWS / 256), dim3(256), 0, stream>>>(x2, n2);

    dim3 grid(TILES, TILES);   // 32 x 32 blocks of 128x128
    dim3 block(256);
    mmd_tile_kernel<<<grid, block, 0, stream>>>(x1, x1, n1, n1,
                                                partials + 0 * NPART, 1);
    mmd_tile_kernel<<<grid, block, 0, stream>>>(x2, x2, n2, n2,
                                                partials + 1 * NPART, 1);
    mmd_tile_kernel<<<grid, block, 0, stream>>>(x1, x2, n1, n2,
                                                partials + 2 * NPART, 0);

    mmd_finalize_kernel<<<1, 256, 0, stream>>>(partials, out);
}